// GATNet_5652176962007
// MI455X (gfx1250) — compile-verified
//
#include <hip/hip_runtime.h>
#include <math.h>

#define NNODES   50000
#define NEDGES   800000
#define ETOT     (NNODES + NEDGES)
#define FIN      128
#define CH       64
#define NLAYERS  6
#define SLOPE    0.2f
#define NTILES   (NNODES / 16)        // 3125 row tiles, exact

typedef __attribute__((ext_vector_type(2))) float v2f;
typedef __attribute__((ext_vector_type(8))) float v8f;

// ---------------- order-preserving float<->uint encoding for atomicMax ----
__device__ __forceinline__ unsigned enc_f(float f) {
    unsigned u = __float_as_uint(f);
    return (u & 0x80000000u) ? ~u : (u | 0x80000000u);
}
__device__ __forceinline__ float dec_f(unsigned e) {
    return (e & 0x80000000u) ? __uint_as_float(e & 0x7FFFFFFFu)
                             : __uint_as_float(~e);
}
__device__ __forceinline__ float lrelu(float v) { return v > 0.f ? v : SLOPE * v; }

// ---------------- fused dual GEMM:  g = hin @ W,  r = hin @ Wr ------------
// 256-thread block = 8 waves; W/Wr staged once per block into LDS with the
// two K-rows of each WMMA fragment interleaved so the per-step B read is a
// single ds_load_b64. Each wave owns one 16-row strip, 4 col tiles, and
// V_WMMA_F32_16X16X4_F32 accumulators (fp32 matrix-core path, wave32).
template <int K>
__global__ __launch_bounds__(256)
void gat_gemm_wmma(const float* __restrict__ hin,
                   const float* __restrict__ W,
                   const float* __restrict__ Wr,
                   float* __restrict__ g,
                   float* __restrict__ r) {
    extern __shared__ float lds[];            // [2][K*CH] pair-interleaved
    float* sW  = lds;
    float* sWr = lds + K * CH;

    // ---- stage: lds[(kk>>1)*128 + c*2 + (kk&1)] = W[kk*64 + c] ----
    for (int idx = threadIdx.x * 4; idx < K * CH; idx += 256 * 4) {
        const int kk = idx >> 6;
        const int c  = idx & 63;
        const float4 w  = *(const float4*)(W  + idx);
        const float4 wr = *(const float4*)(Wr + idx);
        const int base = ((kk >> 1) << 7) + (kk & 1);
        sW [base + ((c + 0) << 1)] = w.x;
        sW [base + ((c + 1) << 1)] = w.y;
        sW [base + ((c + 2) << 1)] = w.z;
        sW [base + ((c + 3) << 1)] = w.w;
        sWr[base + ((c + 0) << 1)] = wr.x;
        sWr[base + ((c + 1) << 1)] = wr.y;
        sWr[base + ((c + 2) << 1)] = wr.z;
        sWr[base + ((c + 3) << 1)] = wr.w;
    }
    __syncthreads();

    const int wid  = threadIdx.x >> 5;
    const int lane = threadIdx.x & 31;
    const int wt   = blockIdx.x * 8 + wid;    // row tile handled by this wave
    if (wt >= NTILES) return;

    const int half = lane >> 4;               // K sub-group: 0 -> k+{0,1}, 1 -> k+{2,3}
    const int lid  = lane & 15;
    const int Mb   = wt * 16;

    v8f cg[4] = {v8f{}, v8f{}, v8f{}, v8f{}};
    v8f cr[4] = {v8f{}, v8f{}, v8f{}, v8f{}};

    const float* arow = hin + (size_t)(Mb + lid) * K;

    for (int k = 0; k < K; k += 4) {
        const int ka = k + half * 2;          // even
        const v2f a  = *(const v2f*)(arow + ka);
        const float* bwp  = sW  + ((ka >> 1) << 7);
        const float* bwrp = sWr + ((ka >> 1) << 7);
#pragma unroll
        for (int t = 0; t < 4; ++t) {
            const int col = t * 16 + lid;
            const v2f b  = *(const v2f*)(bwp  + (col << 1));
            const v2f br = *(const v2f*)(bwrp + (col << 1));
            cg[t] = __builtin_amdgcn_wmma_f32_16x16x4_f32(
                        false, a, false, b,  (short)0, cg[t], false, false);
            cr[t] = __builtin_amdgcn_wmma_f32_16x16x4_f32(
                        false, a, false, br, (short)0, cr[t], false, false);
        }
    }

#pragma unroll
    for (int t = 0; t < 4; ++t) {
#pragma unroll
        for (int i = 0; i < 8; ++i) {
            const int row = Mb + i + half * 8;   // C/D layout: VGPR i -> M=i / M=i+8
            const int col = t * 16 + lid;
            g[(size_t)row * CH + col] = cg[t][i];
            r[(size_t)row * CH + col] = cr[t][i];
        }
    }
}

// ---------------- wave-per-node attention logits: al = g.a_src, ar = g.a_dst
__global__ __launch_bounds__(256)
void gat_attn_dots(const float* __restrict__ g,
                   const float* __restrict__ as,
                   const float* __restrict__ ad,
                   float* __restrict__ al, float* __restrict__ ar, int n) {
    const int wave = (int)((blockIdx.x * blockDim.x + threadIdx.x) >> 5);
    const int lane = threadIdx.x & 31;
    if (wave >= n) return;
    const float* row = g + (size_t)wave * CH;
    float va = row[lane] * as[lane] + row[lane + 32] * as[lane + 32];
    float vd = row[lane] * ad[lane] + row[lane + 32] * ad[lane + 32];
#pragma unroll
    for (int off = 16; off > 0; off >>= 1) {
        va += __shfl_xor(va, off, 32);
        vd += __shfl_xor(vd, off, 32);
    }
    if (lane == 0) { al[wave] = va; ar[wave] = vd; }
}

// ---------------- final-layer projections (CH -> 1) -----------------------
__global__ __launch_bounds__(256)
void gat_final_dots(const float* __restrict__ h,
                    const float* __restrict__ Wo,
                    const float* __restrict__ lWo,
                    const float* __restrict__ aso,
                    const float* __restrict__ ado,
                    float* __restrict__ go, float* __restrict__ ro,
                    float* __restrict__ al, float* __restrict__ ar, int n) {
    const int wave = (int)((blockIdx.x * blockDim.x + threadIdx.x) >> 5);
    const int lane = threadIdx.x & 31;
    if (wave >= n) return;
    const float* row = h + (size_t)wave * CH;
    float vg = row[lane] * Wo[lane]  + row[lane + 32] * Wo[lane + 32];
    float vr = row[lane] * lWo[lane] + row[lane + 32] * lWo[lane + 32];
#pragma unroll
    for (int off = 16; off > 0; off >>= 1) {
        vg += __shfl_xor(vg, off, 32);
        vr += __shfl_xor(vr, off, 32);
    }
    if (lane == 0) {
        go[wave] = vg;
        ro[wave] = vr;
        al[wave] = vg * aso[0];
        ar[wave] = vg * ado[0];
    }
}

// ---------------- zero scratch --------------------------------------------
__global__ void gat_zero(float* p, int n) {
    int t = blockIdx.x * blockDim.x + threadIdx.x;
    if (t < n) p[t] = 0.f;
}

// ---------------- edge pass 1: per-dst segment max ------------------------
__global__ void gat_edge_max(const int* __restrict__ ei,
                             const float* __restrict__ al,
                             const float* __restrict__ ar,
                             unsigned* __restrict__ mb) {
    int e = blockIdx.x * blockDim.x + threadIdx.x;
    if (e >= ETOT) return;
    int s = (e < NEDGES) ? ei[e] : e - NEDGES;
    int d = (e < NEDGES) ? ei[NEDGES + e] : e - NEDGES;
    atomicMax(mb + d, enc_f(lrelu(al[s] + ar[d])));
}

// ---------------- edge pass 2: per-dst exp-sum ----------------------------
__global__ void gat_edge_sum(const int* __restrict__ ei,
                             const float* __restrict__ al,
                             const float* __restrict__ ar,
                             const unsigned* __restrict__ mb,
                             float* __restrict__ ssum) {
    int e = blockIdx.x * blockDim.x + threadIdx.x;
    if (e >= ETOT) return;
    int s = (e < NEDGES) ? ei[e] : e - NEDGES;
    int d = (e < NEDGES) ? ei[NEDGES + e] : e - NEDGES;
    float ex = __expf(lrelu(al[s] + ar[d]) - dec_f(mb[d]));
    atomicAdd(ssum + d, ex);
}

// ---------------- edge pass 3: alpha-weighted scatter (64 channels) -------
// 4 threads per edge; each handles 16 channels with float4 gathers.
__global__ void gat_edge_scatter64(const int* __restrict__ ei,
                                   const float* __restrict__ al,
                                   const float* __restrict__ ar,
                                   const unsigned* __restrict__ mb,
                                   const float* __restrict__ ssum,
                                   const float* __restrict__ g,
                                   float* __restrict__ acc) {
    int t = blockIdx.x * blockDim.x + threadIdx.x;
    int e = t >> 2, q = t & 3;
    if (e >= ETOT) return;
    int s = (e < NEDGES) ? ei[e] : e - NEDGES;
    int d = (e < NEDGES) ? ei[NEDGES + e] : e - NEDGES;
    float alpha = __expf(lrelu(al[s] + ar[d]) - dec_f(mb[d])) / ssum[d];
    const float4* gs = (const float4*)(g + (size_t)s * CH + q * 16);
    float* ap = acc + (size_t)d * CH + q * 16;
#pragma unroll
    for (int j = 0; j < 4; ++j) {
        float4 v = gs[j];
        atomicAdd(ap + j * 4 + 0, alpha * v.x);
        atomicAdd(ap + j * 4 + 1, alpha * v.y);
        atomicAdd(ap + j * 4 + 2, alpha * v.z);
        atomicAdd(ap + j * 4 + 3, alpha * v.w);
    }
}

// ---------------- edge pass 3 (final layer, 1 channel) --------------------
__global__ void gat_edge_scatter1(const int* __restrict__ ei,
                                  const float* __restrict__ al,
                                  const float* __restrict__ ar,
                                  const unsigned* __restrict__ mb,
                                  const float* __restrict__ ssum,
                                  const float* __restrict__ go,
                                  float* __restrict__ acc) {
    int e = blockIdx.x * blockDim.x + threadIdx.x;
    if (e >= ETOT) return;
    int s = (e < NEDGES) ? ei[e] : e - NEDGES;
    int d = (e < NEDGES) ? ei[NEDGES + e] : e - NEDGES;
    float alpha = __expf(lrelu(al[s] + ar[d]) - dec_f(mb[d])) / ssum[d];
    atomicAdd(acc + d, alpha * go[s]);
}

// ---------------- epilogues ------------------------------------------------
__global__ void gat_finalize_elu(const float* __restrict__ acc,
                                 const float* __restrict__ r,
                                 const float* __restrict__ bias,
                                 const float* __restrict__ lb,
                                 float* __restrict__ hout) {
    int t = blockIdx.x * blockDim.x + threadIdx.x;
    if (t >= NNODES * CH) return;
    int c = t & (CH - 1);
    float v = acc[t] + bias[c] + r[t] + lb[c];
    hout[t] = v > 0.f ? v : expm1f(v);
}

__global__ void gat_finalize_out(const float* __restrict__ acc,
                                 const float* __restrict__ ro,
                                 const float* __restrict__ bo,
                                 const float* __restrict__ lbo,
                                 float* __restrict__ out) {
    int i = blockIdx.x * blockDim.x + threadIdx.x;
    if (i >= NNODES) return;
    out[i] = acc[i] + bo[0] + ro[i] + lbo[0];
}

// ---------------- host orchestration --------------------------------------
static void run_edge_layer(const int* ei, float* al, float* ar, unsigned* mb,
                           float* ssum, hipStream_t stream) {
    dim3 eb((ETOT + 255) / 256), tb(256);
    gat_edge_max<<<eb, tb, 0, stream>>>(ei, al, ar, mb);
    gat_edge_sum<<<eb, tb, 0, stream>>>(ei, al, ar, mb, ssum);
}

extern "C" void kernel_launch(void* const* d_in, const int* in_sizes, int n_in,
                              void* d_out, int out_size, void* d_ws, size_t ws_size,
                              hipStream_t stream) {
    (void)in_sizes; (void)n_in; (void)out_size; (void)ws_size;
    const float* x   = (const float*)d_in[0];
    const int*   ei  = (const int*)  d_in[1];
    const float* Wi  = (const float*)d_in[2];
    const float* asi = (const float*)d_in[3];
    const float* adi = (const float*)d_in[4];
    const float* bi  = (const float*)d_in[5];
    const float* lWi = (const float*)d_in[6];
    const float* lbi = (const float*)d_in[7];
    const float* Wh  = (const float*)d_in[8];
    const float* ash = (const float*)d_in[9];
    const float* adh = (const float*)d_in[10];
    const float* bh  = (const float*)d_in[11];
    const float* lWh = (const float*)d_in[12];
    const float* lbh = (const float*)d_in[13];
    const float* Wo  = (const float*)d_in[14];
    const float* aso = (const float*)d_in[15];
    const float* ado = (const float*)d_in[16];
    const float* bo  = (const float*)d_in[17];
    const float* lWo = (const float*)d_in[18];
    const float* lbo = (const float*)d_in[19];

    const size_t NC = (size_t)NNODES * CH;
    float*    ws   = (float*)d_ws;
    float*    hA   = ws;
    float*    hB   = ws + NC;
    float*    g    = ws + 2 * NC;
    float*    r    = ws + 3 * NC;
    float*    acc  = ws + 4 * NC;                 // acc | mb | ssum contiguous
    unsigned* mb   = (unsigned*)(ws + 5 * NC);
    float*    ssum = ws + 5 * NC + NNODES;
    float*    al   = ws + 5 * NC + 2 * (size_t)NNODES;
    float*    ar   = ws + 5 * NC + 3 * (size_t)NNODES;

    const dim3 gemmGrid((NTILES + 7) / 8), gemmBlk(256);
    const size_t ldsIn  = (size_t)FIN * CH * 2 * sizeof(float);   // 64 KB
    const size_t ldsHid = (size_t)CH  * CH * 2 * sizeof(float);   // 32 KB
    const dim3 dotGrid((NNODES * 32 + 255) / 256), blk256(256);
    const dim3 zeroGrid((int)((NC + 2 * NNODES + 255) / 256));
    const dim3 sc64Grid((ETOT * 4 + 255) / 256);
    const dim3 finGrid((int)((NC + 255) / 256));
    const dim3 edgeGrid((ETOT + 255) / 256);
    const dim3 nodeGrid((NNODES + 255) / 256);

    // ---- input layer: FIN=128 -> CH=64 ----
    gat_gemm_wmma<FIN><<<gemmGrid, gemmBlk, ldsIn, stream>>>(x, Wi, lWi, g, r);
    gat_attn_dots<<<dotGrid, blk256, 0, stream>>>(g, asi, adi, al, ar, NNODES);
    gat_zero<<<zeroGrid, blk256, 0, stream>>>(acc, (int)(NC + 2 * NNODES));
    run_edge_layer(ei, al, ar, mb, ssum, stream);
    gat_edge_scatter64<<<sc64Grid, blk256, 0, stream>>>(ei, al, ar, mb, ssum, g, acc);
    gat_finalize_elu<<<finGrid, blk256, 0, stream>>>(acc, r, bi, lbi, hA);

    // ---- hidden layers: CH -> CH ----
    float* hin = hA;
    float* hout = hB;
    for (int i = 0; i < NLAYERS; ++i) {
        const float* W_i  = Wh  + (size_t)i * CH * CH;
        const float* lW_i = lWh + (size_t)i * CH * CH;
        gat_gemm_wmma<CH><<<gemmGrid, gemmBlk, ldsHid, stream>>>(hin, W_i, lW_i, g, r);
        gat_attn_dots<<<dotGrid, blk256, 0, stream>>>(g, ash + i * CH, adh + i * CH,
                                                      al, ar, NNODES);
        gat_zero<<<zeroGrid, blk256, 0, stream>>>(acc, (int)(NC + 2 * NNODES));
        run_edge_layer(ei, al, ar, mb, ssum, stream);
        gat_edge_scatter64<<<sc64Grid, blk256, 0, stream>>>(ei, al, ar, mb, ssum, g, acc);
        gat_finalize_elu<<<finGrid, blk256, 0, stream>>>(acc, r, bh + i * CH,
                                                         lbh + i * CH, hout);
        float* tmp = hin; hin = hout; hout = tmp;
    }

    // ---- output layer: CH -> 1 (concat=False, H=1 => mean is identity) ----
    gat_final_dots<<<dotGrid, blk256, 0, stream>>>(hin, Wo, lWo, aso, ado,
                                                   g, r, al, ar, NNODES);
    gat_zero<<<zeroGrid, blk256, 0, stream>>>(acc, (int)(NC + 2 * NNODES));
    run_edge_layer(ei, al, ar, mb, ssum, stream);
    gat_edge_scatter1<<<edgeGrid, blk256, 0, stream>>>(ei, al, ar, mb, ssum, g, acc);
    gat_finalize_out<<<nodeGrid, blk256, 0, stream>>>(acc, r, bo, lbo, (float*)d_out);
}